// RepresentativeVectors_57372173140292
// MI455X (gfx1250) — compile-verified
//
#include <hip/hip_runtime.h>
#include <hip/hip_bf16.h>
#include <math.h>

// Problem constants (from reference): x: (16, 512, 56, 56) fp32
#define BB 16
#define CC 512
#define HH 56
#define WW 56
#define NN (HH * WW)        // 3136
#define NBVEC 16
#define SIGMA_INV 0.05f     // 1/20
#define EPS_C 1e-12f
#define K2_BLOCKS_PER_B 49  // 49 blocks * 4 waves * 16 rows = 3136 = NN

typedef float v2f __attribute__((ext_vector_type(2)));
typedef float v8f __attribute__((ext_vector_type(8)));

// ---------------------------------------------------------------------------
// k0: xsq[b,n] = sum_c x[b,c,n]^2 ; also zero the selectedPos output slab.
// grid = 196 blocks x 256 (exactly B*N threads); coalesced over n.
// ---------------------------------------------------------------------------
__global__ void k0_xsq_init(const float* __restrict__ x,
                            float* __restrict__ xsq,
                            float* __restrict__ posOut) {
  int gid = blockIdx.x * blockDim.x + threadIdx.x;     // 0 .. B*N-1
  int b = gid / NN;
  int n = gid - b * NN;
  const float* xb = x + (size_t)b * CC * NN;
  float s = 0.f;
#pragma unroll 8
  for (int c = 0; c < CC; ++c) {
    float t = xb[(size_t)c * NN + n];
    s += t * t;
  }
  xsq[gid] = s;
  posOut[gid] = 0.f;
}

// ---------------------------------------------------------------------------
// k1: per batch: pick index (N/2 at iter 0, else first-occurrence argmax of
// score), pos += 1, gather v[b,:] = x[b,:,ind], vsq[b] = ||v||^2.
// grid = B blocks x 256 threads. Deterministic tree reductions.
// ---------------------------------------------------------------------------
__global__ void k1_select(const float* __restrict__ x,
                          const float* __restrict__ score,
                          float* __restrict__ v,
                          float* __restrict__ vsq,
                          float* __restrict__ posOut,
                          int iter) {
  __shared__ float smax[256];
  __shared__ int   sidx[256];
  __shared__ float sred[256];
  const int b = blockIdx.x;
  const int t = threadIdx.x;

  int best_i = NN / 2;
  if (iter != 0) {
    const float* sc = score + b * NN;
    float bm = -INFINITY;
    int bi = 0x7fffffff;
    for (int n = t; n < NN; n += 256) {
      float val = sc[n];
      if (val > bm) { bm = val; bi = n; }   // strict > keeps first occurrence
    }
    smax[t] = bm; sidx[t] = bi;
    __syncthreads();
    for (int s = 128; s > 0; s >>= 1) {
      if (t < s) {
        float om = smax[t + s]; int oi = sidx[t + s];
        if (om > smax[t] || (om == smax[t] && oi < sidx[t])) {
          smax[t] = om; sidx[t] = oi;
        }
      }
      __syncthreads();
    }
    best_i = sidx[0];
  }

  if (t == 0) {
    float* p = posOut + b * NN + best_i;
    *p = *p + 1.0f;
  }

  const float* xb = x + (size_t)b * CC * NN;
  float vs = 0.f;
  for (int c = t; c < CC; c += 256) {
    float val = xb[(size_t)c * NN + best_i];
    v[b * CC + c] = val;
    vs += val * val;
  }
  sred[t] = vs;
  __syncthreads();
  for (int s = 128; s > 0; s >>= 1) {
    if (t < s) sred[t] += sred[t + s];
    __syncthreads();
  }
  if (t == 0) vsq[b] = sred[0];
}

// ---------------------------------------------------------------------------
// k2: WMMA f32 16x16x4 GEMV pass.
//   dot[b,n] = sum_c x[b,c,n] * v[b,c]
//   sim = exp(-sqrt(max(xsq - 2dot + vsq, eps))/20)  -> simList output
//   score update; per-block partial sum of sim -> partials.
// A-matrix: 16 pixel rows (n0..n0+15) x 4 channels. Per ISA layout:
//   lanes 0-15 hold (M=lane, K=0 in vgpr0, K=1 in vgpr1),
//   lanes 16-31 hold (M=lane-16, K=2 in vgpr0, K=3 in vgpr1).
// B-matrix: v[c0..c0+3] broadcast across all 16 columns (row k -> vgpr/half).
// D column 0 lives in lanes 0 (rows 0-7) and 16 (rows 8-15).
// grid = B*49 blocks x 128 (4 waves); each wave = one 16-row tile, full EXEC.
// ---------------------------------------------------------------------------
__global__ void k2_sim(const float* __restrict__ x,
                       const float* __restrict__ v,
                       const float* __restrict__ vsq,
                       const float* __restrict__ xsq,
                       float* __restrict__ score,
                       float* __restrict__ simOut,
                       float* __restrict__ partials,
                       int iter) {
  __shared__ float dotLds[4 * 16];
  __shared__ float wsum[4];
  const int b    = blockIdx.x / K2_BLOCKS_PER_B;
  const int blk  = blockIdx.x - b * K2_BLOCKS_PER_B;
  const int wave = threadIdx.x >> 5;
  const int lane = threadIdx.x & 31;
  const int lhi  = lane >> 4;    // which half of the wave
  const int lm   = lane & 15;    // row within tile
  const int n0   = (blk * 4 + wave) * 16;

  const float* xb = x + (size_t)b * CC * NN;
  const float* vb = v + b * CC;

  v8f acc = {};
  for (int c0 = 0; c0 < CC; c0 += 4) {
    const int ke = c0 + 2 * lhi;                  // even K handled by this lane
    v2f a, bm;
    a.x  = xb[(size_t)ke * NN + n0 + lm];         // K = ke
    a.y  = xb[(size_t)(ke + 1) * NN + n0 + lm];   // K = ke+1
    bm.x = vb[ke];
    bm.y = vb[ke + 1];
    acc = __builtin_amdgcn_wmma_f32_16x16x4_f32(
        /*neg_a=*/false, a, /*neg_b=*/false, bm,
        /*c_mod=*/(short)0, acc, /*reuse_a=*/false, /*reuse_b=*/false);
  }

  // Extract column 0 of D: lane 0 has rows 0..7, lane 16 has rows 8..15.
  if (lm == 0) {
#pragma unroll
    for (int r = 0; r < 8; ++r) dotLds[wave * 16 + lhi * 8 + r] = acc[r];
  }
  __syncthreads();

  float simv = 0.f;
  if (lane < 16) {
    const int n = n0 + lane;
    const float dot = dotLds[wave * 16 + lane];
    const float d2  = xsq[b * NN + n] - 2.0f * dot + vsq[b];
    simv = expf(-sqrtf(fmaxf(d2, EPS_C)) * SIGMA_INV);
    simOut[((size_t)b * NBVEC + iter) * NN + n] = simv;
    const float prev = score[b * NN + n];
    score[b * NN + n] = (iter == 0) ? (1.f - simv) : (1.f - simv) * prev;
  }
  // wave sum (lanes >=16 contribute 0), then block sum across 4 waves
  for (int off = 16; off > 0; off >>= 1)
    simv += __shfl_down(simv, off, 32);
  if (lane == 0) wsum[wave] = simv;
  __syncthreads();
  if (threadIdx.x == 0)
    partials[b * K2_BLOCKS_PER_B + blk] = wsum[0] + wsum[1] + wsum[2] + wsum[3];
}

// ---------------------------------------------------------------------------
// k3: vec[b,c] = (sum_n x[b,c,n]*sim[b,n]) / simsum[b]
// grid = B*(C/8) blocks x 256 (8 waves); wave w owns channel cTile*8+w,
// lanes stride over n (coalesced, N = 98*32 exactly), shuffle reduce.
// simsum recomputed per block from the 49 partials (fixed order, cheap).
// ---------------------------------------------------------------------------
__global__ void k3_vec(const float* __restrict__ x,
                       const float* __restrict__ simOut,
                       const float* __restrict__ partials,
                       float* __restrict__ vecOut,
                       int iter) {
  const int b    = blockIdx.x / (CC / 8);
  const int ct   = blockIdx.x - b * (CC / 8);
  const int wave = threadIdx.x >> 5;
  const int lane = threadIdx.x & 31;
  const int c    = ct * 8 + wave;

  float simsum = 0.f;
#pragma unroll
  for (int p = 0; p < K2_BLOCKS_PER_B; ++p)
    simsum += partials[b * K2_BLOCKS_PER_B + p];

  const float* xb = x + (size_t)b * CC * NN + (size_t)c * NN;
  const float* sb = simOut + ((size_t)b * NBVEC + iter) * NN;
  float acc = 0.f;
  for (int n = lane; n < NN; n += 32)
    acc += xb[n] * sb[n];
  for (int off = 16; off > 0; off >>= 1)
    acc += __shfl_down(acc, off, 32);
  if (lane == 0)
    vecOut[((size_t)b * NBVEC + iter) * CC + c] = acc / simsum;
}

// ---------------------------------------------------------------------------
extern "C" void kernel_launch(void* const* d_in, const int* in_sizes, int n_in,
                              void* d_out, int out_size, void* d_ws, size_t ws_size,
                              hipStream_t stream) {
  (void)in_sizes; (void)n_in; (void)out_size; (void)ws_size;
  const float* x = (const float*)d_in[0];

  // Output layout (return order): vecList [B,NB,C] | simList [B,NB,N] | pos [B,N]
  float* out    = (float*)d_out;
  float* vecOut = out;
  float* simOut = vecOut + (size_t)BB * NBVEC * CC;
  float* posOut = simOut + (size_t)BB * NBVEC * NN;

  // Workspace layout
  float* score    = (float*)d_ws;                  // B*N
  float* xsq      = score + (size_t)BB * NN;       // B*N
  float* v        = xsq + (size_t)BB * NN;         // B*C
  float* vsq      = v + (size_t)BB * CC;           // B
  float* partials = vsq + BB;                      // B*49

  k0_xsq_init<<<(BB * NN) / 256, 256, 0, stream>>>(x, xsq, posOut);

  for (int i = 0; i < NBVEC; ++i) {
    k1_select<<<BB, 256, 0, stream>>>(x, score, v, vsq, posOut, i);
    k2_sim<<<BB * K2_BLOCKS_PER_B, 128, 0, stream>>>(x, v, vsq, xsq, score,
                                                     simOut, partials, i);
    k3_vec<<<BB * (CC / 8), 256, 0, stream>>>(x, simOut, partials, vecOut, i);
  }
}